// EnhancedFlowLayer_37520834298257
// MI455X (gfx1250) — compile-verified
//
#include <hip/hip_runtime.h>
#include <hip/hip_bf16.h>

typedef __attribute__((ext_vector_type(2))) float v2f;
typedef __attribute__((ext_vector_type(4))) float v4f;
typedef __attribute__((ext_vector_type(8))) float v8f;

#define S_ 256
#define D_ 512
#define P_ 16
#define DD_ 262144   // D*D
#define MAXSEQ_ 4096.0f

__device__ __forceinline__ float gelu_f(float x) {
    return 0.5f * x * (1.0f + erff(x * 0.70710678118654752f));
}
__device__ __forceinline__ float silu_f(float x) {
    return x / (1.0f + expf(-x));
}
__device__ __forceinline__ float sigmoid_f(float x) {
    return 1.0f / (1.0f + expf(-x));
}

// ---------------- K0: LayerNorm + RoPE ----------------
__global__ __launch_bounds__(256) void ln_rope_kernel(
    const float* __restrict__ x, const float* __restrict__ g, const float* __restrict__ b,
    float* __restrict__ xnorm, float* __restrict__ xr)
{
    int s = blockIdx.x, t = threadIdx.x;
    const float* row = x + s * D_;
    float v0 = row[2 * t], v1 = row[2 * t + 1];
    __shared__ float red[256];
    red[t] = v0 + v1; __syncthreads();
    for (int o = 128; o > 0; o >>= 1) { if (t < o) red[t] += red[t + o]; __syncthreads(); }
    float m = red[0] * (1.0f / D_);
    __syncthreads();
    red[t] = v0 * v0 + v1 * v1; __syncthreads();
    for (int o = 128; o > 0; o >>= 1) { if (t < o) red[t] += red[t + o]; __syncthreads(); }
    float var = red[0] * (1.0f / D_) - m * m;
    float rstd = rsqrtf(var + 1e-5f);
    float xn0 = (v0 - m) * rstd * g[2 * t] + b[2 * t];
    float xn1 = (v1 - m) * rstd * g[2 * t + 1] + b[2 * t + 1];
    xnorm[s * D_ + 2 * t] = xn0;
    xnorm[s * D_ + 2 * t + 1] = xn1;
    float invf = powf(10000.0f, -(2.0f * (float)t) / (float)D_);
    float ang = (float)s * invf;
    float sn = sinf(ang), cs = cosf(ang);
    xr[s * D_ + 2 * t]     = xn0 * cs - xn1 * sn;
    xr[s * D_ + 2 * t + 1] = xn1 * cs + xn0 * sn;
}

// ---------------- K1: column means (ctx over S; memory bank over 512) ----------------
__global__ __launch_bounds__(256) void colmean_kernel(
    const float* __restrict__ xr, const float* __restrict__ memory_bank,
    float* __restrict__ ctx, float* __restrict__ membar)
{
    int d = blockIdx.x * 256 + threadIdx.x;
    float a = 0.0f;
    for (int s = 0; s < S_; ++s) a += xr[s * D_ + d];
    ctx[d] = a * (1.0f / S_);
    float msum = 0.0f;
    for (int r = 0; r < 512; ++r) msum += memory_bank[r * D_ + d];
    membar[d] = msum * (1.0f / 512.0f);
}

// ---------------- K2: per-token selector + intensity MLPs ----------------
__global__ __launch_bounds__(256) void selector_kernel(
    const float* __restrict__ xr, const float* __restrict__ ctx,
    const float* __restrict__ sel_w1, const float* __restrict__ sel_b1,
    const float* __restrict__ sel_w2, const float* __restrict__ sel_b2,
    const float* __restrict__ int_w1, const float* __restrict__ int_b1,
    const float* __restrict__ int_w2, const float* __restrict__ int_b2,
    float* __restrict__ pw_out, float* __restrict__ inten_out)
{
    int s = blockIdx.x, t = threadIdx.x;
    __shared__ float comb[1024];
    __shared__ float red[256];
    __shared__ float h1[32];
    __shared__ float hi[64];
    __shared__ float lg[16];
    comb[t] = xr[s * D_ + t];
    comb[t + 256] = xr[s * D_ + t + 256];
    comb[512 + t] = ctx[t];
    comb[768 + t] = ctx[t + 256];
    __syncthreads();

    // h1 = gelu(comb @ sel_w1 + sel_b1)  (1024 -> 32)
    {
        int j = t & 31, c = t >> 5;           // 8 chunks x 128
        float p = 0.0f;
        for (int k = c * 128; k < c * 128 + 128; ++k) p += comb[k] * sel_w1[k * 32 + j];
        red[t] = p; __syncthreads();
        if (t < 32) {
            float a = sel_b1[t];
            for (int c2 = 0; c2 < 8; ++c2) a += red[c2 * 32 + t];
            h1[t] = gelu_f(a);
        }
        __syncthreads();
    }
    // logits = h1 @ sel_w2 + sel_b2 (32 -> 16), softmax
    if (t < 16) {
        float a = sel_b2[t];
        for (int j = 0; j < 32; ++j) a += h1[j] * sel_w2[j * 16 + t];
        lg[t] = a;
    }
    __syncthreads();
    if (t == 0) {
        float mx = lg[0];
        for (int p = 1; p < 16; ++p) mx = fmaxf(mx, lg[p]);
        float sm = 0.0f;
        float e[16];
        for (int p = 0; p < 16; ++p) { e[p] = expf(lg[p] - mx); sm += e[p]; }
        float inv = 1.0f / sm;
        for (int p = 0; p < 16; ++p) pw_out[s * P_ + p] = e[p] * inv;
    }
    __syncthreads();
    // hi = gelu(comb @ int_w1 + int_b1) (1024 -> 64)
    {
        int j = t & 63, c = t >> 6;           // 4 chunks x 256
        float p = 0.0f;
        for (int k = c * 256; k < c * 256 + 256; ++k) p += comb[k] * int_w1[k * 64 + j];
        red[t] = p; __syncthreads();
        if (t < 64) {
            float a = int_b1[t];
            for (int c2 = 0; c2 < 4; ++c2) a += red[c2 * 64 + t];
            hi[t] = gelu_f(a);
        }
        __syncthreads();
    }
    if (t == 0) {
        float a = int_b2[0];
        for (int j = 0; j < 64; ++j) a += hi[j] * int_w2[j];
        inten_out[s] = sigmoid_f(a);
    }
}

// ---------------- K3: adaptive window -> kk ----------------
__global__ __launch_bounds__(64) void window_kernel(
    const float* __restrict__ ctx,
    const float* __restrict__ win_w1, const float* __restrict__ win_b1,
    const float* __restrict__ win_w2, const float* __restrict__ win_b2,
    int* __restrict__ kk_out)
{
    int t = threadIdx.x;
    __shared__ float hw[64];
    float a = win_b1[t];
    for (int k = 0; k < D_; ++k) a += ctx[k] * win_w1[k * 64 + t];
    hw[t] = gelu_f(a);
    __syncthreads();
    if (t == 0) {
        float l = win_b2[0];
        for (int j = 0; j < 64; ++j) l += hw[j] * win_w2[j];
        float win = sigmoid_f(l) * (MAXSEQ_ - 256.0f) + 256.0f;
        float asp = 0.1f * (win / MAXSEQ_);
        int kk = (int)((float)DD_ * asp);
        if (kk < 1) kk = 1;
        *kk_out = kk;
    }
}

// ---------------- K4: exact per-token kk-th largest |flow| via 3-pass radix select ----------------
// Vectorized: each lane combines 4 consecutive idx per pattern via b128 loads.
__global__ __launch_bounds__(256) void radix_select_kernel(
    const float* __restrict__ fp, const float* __restrict__ pw_in,
    const float* __restrict__ inten_in, const int* __restrict__ kk_in,
    float* __restrict__ th_out)
{
    __shared__ unsigned hist[2048];
    __shared__ float pwsh[16];
    __shared__ unsigned sprefix, shimask;
    __shared__ int sremaining;
    int s = blockIdx.x, t = threadIdx.x;
    if (t < 16) pwsh[t] = pw_in[s * P_ + t];
    if (t == 0) { sprefix = 0u; shimask = 0u; sremaining = *kk_in; }
    float inten = inten_in[s];
    __syncthreads();
    float wv[16];
#pragma unroll
    for (int p = 0; p < 16; ++p) wv[p] = pwsh[p];

    const int shifts[3] = {20, 9, 0};
    const unsigned nbs[3] = {2048u, 2048u, 512u};
    for (int pass = 0; pass < 3; ++pass) {
        int shift = shifts[pass];
        unsigned nb = nbs[pass];
        for (int i = t; i < 2048; i += 256) hist[i] = 0u;
        __syncthreads();
        unsigned prefix = sprefix, himask = shimask;
        for (int idx0 = t * 4; idx0 < DD_; idx0 += 1024) {
            v4f f4 = {0.0f, 0.0f, 0.0f, 0.0f};
#pragma unroll
            for (int p = 0; p < 16; ++p) {
                v4f d = *(const v4f*)(fp + p * DD_ + idx0);
                f4 += d * wv[p];
            }
#pragma unroll
            for (int e = 0; e < 4; ++e) {
                unsigned v = __float_as_uint(fabsf(f4[e] * inten));
                if ((v & himask) == prefix) atomicAdd(&hist[(v >> shift) & (nb - 1u)], 1u);
            }
        }
        __syncthreads();
        if (t == 0) {
            int rem = sremaining;
            unsigned cum = 0u, sel = 0u;
            for (int bb = (int)nb - 1; bb >= 0; --bb) {
                cum += hist[bb];
                if ((int)cum >= rem) { sel = (unsigned)bb; sremaining = rem - (int)(cum - hist[bb]); break; }
            }
            sprefix = prefix | (sel << shift);
            shimask = himask | ((nb - 1u) << shift);
        }
        __syncthreads();
    }
    if (t == 0) th_out[s] = __uint_as_float(sprefix);
}

// ---------------- K5: masked flow matvec (vectorized b128 pattern reads) ----------------
__global__ __launch_bounds__(256) void flow_matvec_kernel(
    const float* __restrict__ fp, const float* __restrict__ pw_in,
    const float* __restrict__ inten_in, const float* __restrict__ th_in,
    const float* __restrict__ xnorm, float* __restrict__ flow_out)
{
    __shared__ float xn[512];
    __shared__ float pwsh[16];
    int s = blockIdx.x, t = threadIdx.x;
    xn[t] = xnorm[s * D_ + t];
    xn[t + 256] = xnorm[s * D_ + t + 256];
    if (t < 16) pwsh[t] = pw_in[s * P_ + t];
    __syncthreads();
    float inten = inten_in[s], th = th_in[s];
    float wv[16];
#pragma unroll
    for (int p = 0; p < 16; ++p) wv[p] = pwsh[p];
    int wave = t >> 5, lane = t & 31;
    for (int i = wave; i < D_; i += 8) {
        float acc = 0.0f;
        const float* base = fp + i * D_;
#pragma unroll
        for (int jb = 0; jb < 4; ++jb) {
            int j0 = jb * 128 + lane * 4;
            v4f f4 = {0.0f, 0.0f, 0.0f, 0.0f};
#pragma unroll
            for (int p = 0; p < 16; ++p) {
                v4f d = *(const v4f*)(base + p * DD_ + j0);
                f4 += d * wv[p];
            }
#pragma unroll
            for (int e = 0; e < 4; ++e) {
                float f = f4[e] * inten;
                if (fabsf(f) >= th) acc += f * xn[j0 + e];
            }
        }
        for (int m = 16; m >= 1; m >>= 1) acc += __shfl_xor(acc, m, 32);
        if (lane == 0) flow_out[s * D_ + i] = acc;
    }
}

// ---------------- K6: co = LN2(x + flow_out); comb = [co | membar] ----------------
__global__ __launch_bounds__(256) void add_ln_comb_kernel(
    const float* __restrict__ x, const float* __restrict__ flow_out,
    const float* __restrict__ g, const float* __restrict__ b,
    const float* __restrict__ membar, float* __restrict__ co, float* __restrict__ comb)
{
    int s = blockIdx.x, t = threadIdx.x;
    float v0 = x[s * D_ + t] + flow_out[s * D_ + t];
    float v1 = x[s * D_ + t + 256] + flow_out[s * D_ + t + 256];
    __shared__ float red[256];
    red[t] = v0 + v1; __syncthreads();
    for (int o = 128; o > 0; o >>= 1) { if (t < o) red[t] += red[t + o]; __syncthreads(); }
    float m = red[0] * (1.0f / D_);
    __syncthreads();
    red[t] = v0 * v0 + v1 * v1; __syncthreads();
    for (int o = 128; o > 0; o >>= 1) { if (t < o) red[t] += red[t + o]; __syncthreads(); }
    float var = red[0] * (1.0f / D_) - m * m;
    float rstd = rsqrtf(var + 1e-5f);
    float c0 = (v0 - m) * rstd * g[t] + b[t];
    float c1 = (v1 - m) * rstd * g[t + 256] + b[t + 256];
    co[s * D_ + t] = c0;
    co[s * D_ + t + 256] = c1;
    comb[s * 1024 + t] = c0;
    comb[s * 1024 + t + 256] = c1;
    comb[s * 1024 + 512 + t] = membar[t];
    comb[s * 1024 + 768 + t] = membar[t + 256];
}

// ---------------- WMMA fp32 GEMM: C[M,N] = epilogue(A[M,K] @ B[K,N] + bias) ----------------
// mode 0: none; mode 1: silu; mode 2: += aux (residual)
// Each wave computes one 16x16 tile using v_wmma_f32_16x16x4_f32 (exact fp32).
__global__ __launch_bounds__(256) void wmma_gemm_f32(
    const float* __restrict__ A, const float* __restrict__ B,
    const float* __restrict__ bias, const float* __restrict__ aux,
    float* __restrict__ C, int M, int N, int K, int mode)
{
    int wave = threadIdx.x >> 5;
    int lane = threadIdx.x & 31;
    int laneM = lane & 15;
    int half = lane >> 4;
    int n0 = (blockIdx.x * 8 + wave) * 16;
    int m0 = blockIdx.y * 16;

    v8f acc = {};
    const float* Arow = A + (m0 + laneM) * K;
    const float* Bcol = B + n0 + laneM;
    for (int k16 = 0; k16 < K; k16 += 16) {
        __builtin_prefetch(Arow + k16 + 64, 0, 1);   // global_prefetch_b8: next A chunk
#pragma unroll
        for (int ku = 0; ku < 16; ku += 4) {
            int ka = k16 + ku + 2 * half;
            v2f a = *(const v2f*)(Arow + ka);        // A[row][ka], A[row][ka+1]
            v2f bb;
            bb.x = Bcol[ka * N];                     // B[ka][col]
            bb.y = Bcol[(ka + 1) * N];               // B[ka+1][col]
            acc = __builtin_amdgcn_wmma_f32_16x16x4_f32(
                false, a, false, bb, (short)0, acc, false, false);
        }
    }
    int col = n0 + laneM;
    float bv = bias[col];
#pragma unroll
    for (int r = 0; r < 8; ++r) {
        int row = m0 + r + 8 * half;
        float v = acc[r] + bv;
        if (mode == 1) v = silu_f(v);
        else if (mode == 2) v += aux[row * N + col];
        C[row * N + col] = v;
    }
}

// ---------------- gate/val elementwise for FFN ----------------
__global__ __launch_bounds__(256) void gateval_kernel(
    const float* __restrict__ ff, float* __restrict__ hg)
{
    int idx = blockIdx.x * 256 + threadIdx.x;   // 256 * 2048 total
    int s = idx >> 11, j = idx & 2047;
    float gt = ff[s * 4096 + j];
    float vl = ff[s * 4096 + 2048 + j];
    hg[idx] = silu_f(gt) * vl;
}

extern "C" void kernel_launch(void* const* d_in, const int* in_sizes, int n_in,
                              void* d_out, int out_size, void* d_ws, size_t ws_size,
                              hipStream_t stream) {
    const float* x        = (const float*)d_in[0];
    const float* fp       = (const float*)d_in[1];
    const float* sel_w1   = (const float*)d_in[2];
    const float* sel_b1   = (const float*)d_in[3];
    const float* sel_w2   = (const float*)d_in[4];
    const float* sel_b2   = (const float*)d_in[5];
    const float* win_w1   = (const float*)d_in[6];
    const float* win_b1   = (const float*)d_in[7];
    const float* win_w2   = (const float*)d_in[8];
    const float* win_b2   = (const float*)d_in[9];
    const float* int_w1   = (const float*)d_in[10];
    const float* int_b1   = (const float*)d_in[11];
    const float* int_w2   = (const float*)d_in[12];
    const float* int_b2   = (const float*)d_in[13];
    const float* mem_w1   = (const float*)d_in[14];
    const float* mem_b1   = (const float*)d_in[15];
    const float* mem_w2   = (const float*)d_in[16];
    const float* mem_b2   = (const float*)d_in[17];
    const float* membank  = (const float*)d_in[18];
    const float* up_w     = (const float*)d_in[19];
    const float* up_b     = (const float*)d_in[20];
    const float* down_w   = (const float*)d_in[21];
    const float* down_b   = (const float*)d_in[22];
    const float* n1_g     = (const float*)d_in[23];
    const float* n1_b     = (const float*)d_in[24];
    const float* n2_g     = (const float*)d_in[25];
    const float* n2_b     = (const float*)d_in[26];

    float* w = (float*)d_ws;
    float* xnorm    = w; w += S_ * D_;        // 131072
    float* xr       = w; w += S_ * D_;        // 131072
    float* ctx      = w; w += D_;             // 512
    float* membar   = w; w += D_;             // 512
    float* pw       = w; w += S_ * P_;        // 4096
    float* inten    = w; w += S_;             // 256
    float* th       = w; w += S_;             // 256
    int*   kk       = (int*)w; w += 64;       // padded slot
    float* flow_out = w; w += S_ * D_;        // 131072
    float* co       = w; w += S_ * D_;        // 131072
    float* comb     = w; w += S_ * 1024;      // 262144
    float* hbuf     = w; w += S_ * D_;        // 131072
    float* co2      = w; w += S_ * D_;        // 131072
    float* ff       = w; w += S_ * 4096;      // 1048576
    float* hg       = w; w += S_ * 2048;      // 524288

    float* out = (float*)d_out;

    ln_rope_kernel<<<S_, 256, 0, stream>>>(x, n1_g, n1_b, xnorm, xr);
    colmean_kernel<<<2, 256, 0, stream>>>(xr, membank, ctx, membar);
    selector_kernel<<<S_, 256, 0, stream>>>(xr, ctx, sel_w1, sel_b1, sel_w2, sel_b2,
                                            int_w1, int_b1, int_w2, int_b2, pw, inten);
    window_kernel<<<1, 64, 0, stream>>>(ctx, win_w1, win_b1, win_w2, win_b2, kk);
    radix_select_kernel<<<S_, 256, 0, stream>>>(fp, pw, inten, kk, th);
    flow_matvec_kernel<<<S_, 256, 0, stream>>>(fp, pw, inten, th, xnorm, flow_out);
    add_ln_comb_kernel<<<S_, 256, 0, stream>>>(x, flow_out, n2_g, n2_b, membar, co, comb);

    // mem MLP: h = silu(comb @ mem_w1 + b1); co2 = co + (h @ mem_w2 + b2)
    wmma_gemm_f32<<<dim3(512 / 128, S_ / 16), 256, 0, stream>>>(
        comb, mem_w1, mem_b1, nullptr, hbuf, S_, 512, 1024, 1);
    wmma_gemm_f32<<<dim3(512 / 128, S_ / 16), 256, 0, stream>>>(
        hbuf, mem_w2, mem_b2, co, co2, S_, 512, 512, 2);

    // FFN: ff = co2 @ up_w + up_b; hg = silu(gate)*val; out = co2 + hg @ down_w + down_b
    wmma_gemm_f32<<<dim3(4096 / 128, S_ / 16), 256, 0, stream>>>(
        co2, up_w, up_b, nullptr, ff, S_, 4096, 512, 0);
    gateval_kernel<<<(S_ * 2048) / 256, 256, 0, stream>>>(ff, hg);
    wmma_gemm_f32<<<dim3(512 / 128, S_ / 16), 256, 0, stream>>>(
        hg, down_w, down_b, co2, out, S_, 512, 2048, 2);
}